// LLaDASAE_6811818131922
// MI455X (gfx1250) — compile-verified
//
#include <hip/hip_runtime.h>

#define D_MODEL  4096
#define D_SAE    16384
#define K_SPARSE 64
#define N_TOKENS 8192

typedef float v2f  __attribute__((ext_vector_type(2)));
typedef float v8f  __attribute__((ext_vector_type(8)));
typedef unsigned int u32x4 __attribute__((ext_vector_type(4)));
typedef int   i32x4 __attribute__((ext_vector_type(4)));
typedef int   i32x8 __attribute__((ext_vector_type(8)));

// ---------------------------------------------------------------------------
// TDM descriptor (D#) for a 2D f32 tile load: tile_dim0=32 elems (128B rows),
// tile_dim1=64 rows, tensor_dim0_stride = D_MODEL. TDM applies LDS padding:
// pad_interval code 4 (=32 DWORDs) + pad_amount code 3 (=4 DWORDs) -> the
// 36-float LDS row stride used by the WMMA fragment loads (conflict-free).
// CDNA5 ISA: group0 = {flags, lds_addr, global_addr57, type=2},
// group1 = {pad/data_size, tensor dims, tile dims, strides}, groups2/3 = 0
// (2D tensor).  6-arg builtin form (this toolchain's signature).
// ---------------------------------------------------------------------------
__device__ __forceinline__ void tdm_load_tile_64x32(const float* gsrc,
                                                    void* lds_dst)
{
    unsigned long long ga = (unsigned long long)(uintptr_t)gsrc;
    unsigned lds_off = (unsigned)(uintptr_t)lds_dst;   // low 32 bits = LDS byte addr

    u32x4 g0;
    g0[0] = 1u;                                        // count=1, load, user D#
    g0[1] = lds_off;                                   // lds_addr (bytes)
    g0[2] = (unsigned)(ga & 0xFFFFFFFFu);              // global_addr[31:0]
    g0[3] = (unsigned)((ga >> 32) & 0x1FFFFFFu)        // global_addr[56:32]
          | (2u << 30);                                // type = 2 ("image")

    i32x8 g1;
    g1[0] = (2 << 16)                                  // data_size = 4 bytes
          | (1 << 20)                                  // pad_enable
          | (4 << 22)                                  // pad_interval: 32 DWORDs
          | (3 << 25);                                 // pad_amount: 4 DWORDs
    g1[1] = (int)(4096u << 16);                        // tensor_dim0[15:0]=4096 -> bits[63:48]
    g1[2] = (int)((4096u >> 16) | (64u << 16));        // tensor_dim0[31:16] | tensor_dim1[15:0]=64
    g1[3] = (int)(32u << 16);                          // tensor_dim1[31:16]=0 | tile_dim0=32
    g1[4] = (int)64u;                                  // tile_dim1=64 | tile_dim2=0
    g1[5] = (int)D_MODEL;                              // tensor_dim0_stride[31:0]
    g1[6] = 0;                                         // stride0[47:32] | dim1_stride lo (unused, 2D)
    g1[7] = 0;

    i32x4 z4 = {0, 0, 0, 0};
    i32x8 z8 = {0, 0, 0, 0, 0, 0, 0, 0};
    __builtin_amdgcn_tensor_load_to_lds(g0, g1, z4, z4, z8, 0);
}

// ---------------------------------------------------------------------------
// Encoder GEMM: pre[n,f] = dot(x[n,:], W_enc[f,:]) + b_enc[f]   (NT GEMM)
// 64x64 workgroup tile, K in 32-wide blocks, double-buffered LDS filled by
// the Tensor Data Mover (wave 0 issues, TENSORcnt + barrier synchronize),
// compute via V_WMMA_F32_16X16X4_F32 on 8 waves (wave32).
// ---------------------------------------------------------------------------
#define BM 64
#define BN 64
#define BK 32
#define LDSS 36   // 36 floats = 144B row stride (32 data DWORDs + 4 pad DWORDs)

__global__ __launch_bounds__(256) void sae_encoder_wmma(
    const float* __restrict__ x, const float* __restrict__ W_enc,
    const float* __restrict__ b_enc, float* __restrict__ pre)
{
    __shared__ float As[2][BM * LDSS];
    __shared__ float Bs[2][BN * LDSS];

    const int tid  = threadIdx.x;
    const int lane = tid & 31;
    const int wave = tid >> 5;
    const int wm   = wave >> 1;   // 0..3 : 16-row m-block
    const int wn   = wave & 1;    // 0..1 : 32-col n-strip

    const int m0 = blockIdx.y * BM;   // token block
    const int n0 = blockIdx.x * BN;   // feature block

    v8f acc0 = {0.f, 0.f, 0.f, 0.f, 0.f, 0.f, 0.f, 0.f};
    v8f acc1 = {0.f, 0.f, 0.f, 0.f, 0.f, 0.f, 0.f, 0.f};

    // WMMA f32 16x4 A-fragment addressing (ISA layout):
    // lanes 0-15 hold K=kk,kk+1 ; lanes 16-31 hold K=kk+2,kk+3
    const int fm = lane & 15;
    const int fk = (lane >> 4) * 2;

    // Prologue: DMA first K-block into buffer 0 (wave 0 only; EXEC ignored).
    if (wave == 0) {
        tdm_load_tile_64x32(&x[(size_t)m0 * D_MODEL],     (void*)As[0]);
        tdm_load_tile_64x32(&W_enc[(size_t)n0 * D_MODEL], (void*)Bs[0]);
    }

    int cur = 0;
    for (int kb = 0; kb < D_MODEL; kb += BK) {
        if (wave == 0)
            __builtin_amdgcn_s_wait_tensorcnt(0);   // DMA for 'cur' complete
        __syncthreads();                             // publish cur; nxt free

        const int nxt = cur ^ 1;
        if (wave == 0 && (kb + BK) < D_MODEL) {      // DMA next block async
            tdm_load_tile_64x32(&x[(size_t)m0 * D_MODEL + kb + BK],     (void*)As[nxt]);
            tdm_load_tile_64x32(&W_enc[(size_t)n0 * D_MODEL + kb + BK], (void*)Bs[nxt]);
        }

        const float* a_base  = &As[cur][(wm * 16 + fm) * LDSS + fk];
        const float* b0_base = &Bs[cur][(wn * 32 +      fm) * LDSS + fk];
        const float* b1_base = &Bs[cur][(wn * 32 + 16 + fm) * LDSS + fk];
        #pragma unroll
        for (int kk = 0; kk < BK; kk += 4) {
            v2f a  = *(const v2f*)(a_base  + kk);
            v2f b0 = *(const v2f*)(b0_base + kk);
            v2f b1 = *(const v2f*)(b1_base + kk);
            acc0 = __builtin_amdgcn_wmma_f32_16x16x4_f32(
                false, a, false, b0, (short)0, acc0, false, false);
            acc1 = __builtin_amdgcn_wmma_f32_16x16x4_f32(
                false, a, false, b1, (short)0, acc1, false, false);
        }
        cur = nxt;
    }

    // C layout: lanes 0-15 -> M=v, N=lane ; lanes 16-31 -> M=8+v, N=lane-16
    const int cn = lane & 15;
    const int cmBase = m0 + wm * 16 + (lane >> 4) * 8;
    const int col0 = n0 + wn * 32 + cn;
    const int col1 = col0 + 16;
    const float be0 = b_enc[col0];
    const float be1 = b_enc[col1];
    #pragma unroll
    for (int v = 0; v < 8; ++v) {
        const size_t row = (size_t)(cmBase + v);
        pre[row * D_SAE + col0] = acc0[v] + be0;
        pre[row * D_SAE + col1] = acc1[v] + be1;
    }
}

// ---------------------------------------------------------------------------
// Top-K (k=64 of 16384) per token via 4-pass MSB radix select in LDS.
// One workgroup per token; values staged in 64KB dynamic LDS (WGP has 320KB).
// ---------------------------------------------------------------------------
__device__ __forceinline__ unsigned f2key(float f) {
    unsigned u = __float_as_uint(f);
    return (u & 0x80000000u) ? ~u : (u | 0x80000000u);   // monotone order map
}

__global__ __launch_bounds__(256) void sae_topk(
    const float* __restrict__ pre, float* __restrict__ sparse,
    int* __restrict__ topk_idx, float* __restrict__ topk_val)
{
    extern __shared__ char smem[];
    float*    vals = (float*)smem;                               // 16384 f32
    unsigned* hist = (unsigned*)(smem + (size_t)D_SAE * 4);      // 256 bins
    unsigned* ctrl = hist + 256;                                 // 4 words

    const int token = blockIdx.x;
    const int tid   = threadIdx.x;
    const float* prow = pre + (size_t)token * D_SAE;

    for (int j = tid; j < D_SAE; j += 256) vals[j] = prow[j];
    __syncthreads();

    unsigned prefix = 0, pmask = 0, want = K_SPARSE;
    for (int pass = 3; pass >= 0; --pass) {
        hist[tid & 255] = 0;
        __syncthreads();
        const int sh = pass * 8;
        for (int j = tid; j < D_SAE; j += 256) {
            unsigned k = f2key(vals[j]);
            if ((k & pmask) == prefix)
                atomicAdd(&hist[(k >> sh) & 255u], 1u);
        }
        __syncthreads();
        if (tid == 0) {
            unsigned cum = 0; int b = 255;
            for (; b > 0; --b) { cum += hist[b]; if (cum >= want) break; }
            if (cum < want) cum += hist[0];               // b==0 fallthrough
            ctrl[0] = prefix | ((unsigned)b << sh);
            ctrl[1] = want - (cum - hist[b]);             // ties still needed
        }
        __syncthreads();
        prefix = ctrl[0];
        want   = ctrl[1];
        pmask |= (0xFFu << sh);
    }
    const unsigned T = prefix;   // key of k-th largest
    const unsigned E = want;     // #elements equal to T to keep

    if (tid == 0) { ctrl[2] = 0; ctrl[3] = 0; }
    __syncthreads();

    float* srow = sparse + (size_t)token * D_SAE;
    for (int j = tid; j < D_SAE; j += 256) {
        float v = vals[j];
        unsigned k = f2key(v);
        bool keep = (k > T);
        if (k == T) {
            unsigned p = atomicAdd(&ctrl[2], 1u);
            keep = (p < E);
        }
        srow[j] = keep ? v : 0.0f;
        if (keep) {
            unsigned s = atomicAdd(&ctrl[3], 1u);        // exactly 64 total
            topk_idx[token * K_SPARSE + s] = j;
            topk_val[token * K_SPARSE + s] = v;
        }
    }
}

// ---------------------------------------------------------------------------
// W_dec (4096 x 16384) -> W_dec^T (16384 x 4096) so the decoder gather is
// fully coalesced. One-time ~537MB of traffic (~23us at 23.3 TB/s).
// ---------------------------------------------------------------------------
__global__ __launch_bounds__(256) void transpose_wdec(
    const float* __restrict__ Wdec, float* __restrict__ WdecT)
{
    __shared__ float tile[32][33];
    const int tx = threadIdx.x & 31;
    const int ty = threadIdx.x >> 5;          // 0..7
    const int c0 = blockIdx.x * 32;           // along D_SAE
    const int r0 = blockIdx.y * 32;           // along D_MODEL

    #pragma unroll
    for (int r = ty; r < 32; r += 8)
        tile[r][tx] = Wdec[(size_t)(r0 + r) * D_SAE + (c0 + tx)];
    __syncthreads();
    #pragma unroll
    for (int r = ty; r < 32; r += 8)
        WdecT[(size_t)(c0 + r) * D_MODEL + (r0 + tx)] = tile[tx][r];
}

// ---------------------------------------------------------------------------
// Sparse decoder: recon[n,:] = sum_{f active} a[n,f] * WdecT[f,:] + b_dec
// One workgroup per token; float4-coalesced rows; heavy L2 reuse of WdecT;
// explicit prefetch of the next gathered row (global_prefetch_b8).
// ---------------------------------------------------------------------------
__global__ __launch_bounds__(256) void sae_decoder(
    const float* __restrict__ WdecT, const float* __restrict__ b_dec,
    const int* __restrict__ topk_idx, const float* __restrict__ topk_val,
    float* __restrict__ recon)
{
    __shared__ int   sidx[K_SPARSE];
    __shared__ float sval[K_SPARSE];
    const int token = blockIdx.x;
    const int tid   = threadIdx.x;
    if (tid < K_SPARSE) {
        sidx[tid] = topk_idx[token * K_SPARSE + tid];
        sval[tid] = topk_val[token * K_SPARSE + tid];
    }
    __syncthreads();

    const int d0 = tid * 4;                       // + 1024*i, i=0..3
    float4 acc[4];
    #pragma unroll
    for (int i = 0; i < 4; ++i) acc[i] = make_float4(0.f, 0.f, 0.f, 0.f);

    for (int f = 0; f < K_SPARSE; ++f) {
        const float* wrow = WdecT + (size_t)sidx[f] * D_MODEL;
        if (f + 1 < K_SPARSE)                      // hide gather latency
            __builtin_prefetch(WdecT + (size_t)sidx[f + 1] * D_MODEL + d0, 0, 1);
        const float a = sval[f];
        #pragma unroll
        for (int i = 0; i < 4; ++i) {
            float4 w = *(const float4*)&wrow[d0 + 1024 * i];
            acc[i].x += a * w.x; acc[i].y += a * w.y;
            acc[i].z += a * w.z; acc[i].w += a * w.w;
        }
    }

    float* orow = recon + (size_t)token * D_MODEL;
    #pragma unroll
    for (int i = 0; i < 4; ++i) {
        float4 b = *(const float4*)&b_dec[d0 + 1024 * i];
        float4 o = make_float4(acc[i].x + b.x, acc[i].y + b.y,
                               acc[i].z + b.z, acc[i].w + b.w);
        *(float4*)&orow[d0 + 1024 * i] = o;
    }
}

// ---------------------------------------------------------------------------
extern "C" void kernel_launch(void* const* d_in, const int* in_sizes, int n_in,
                              void* d_out, int out_size, void* d_ws, size_t ws_size,
                              hipStream_t stream)
{
    const float* x     = (const float*)d_in[0];
    const float* W_enc = (const float*)d_in[1];
    const float* b_enc = (const float*)d_in[2];
    const float* W_dec = (const float*)d_in[3];
    const float* b_dec = (const float*)d_in[4];

    // d_out = [recon | sparse_acts | pre_acts], all f32
    float* recon  = (float*)d_out;
    float* sparse = recon  + (size_t)N_TOKENS * D_MODEL;
    float* pre    = sparse + (size_t)N_TOKENS * D_SAE;

    // workspace: W_dec^T (268MB) + topk idx/val lists (4MB)
    float* WdecT = (float*)d_ws;
    int*   tidx  = (int*)  ((char*)d_ws + (size_t)D_SAE * D_MODEL * sizeof(float));
    float* tval  = (float*)((char*)tidx + (size_t)N_TOKENS * K_SPARSE * sizeof(int));

    transpose_wdec<<<dim3(D_SAE / 32, D_MODEL / 32), 256, 0, stream>>>(W_dec, WdecT);
    sae_encoder_wmma<<<dim3(D_SAE / BN, N_TOKENS / BM), 256, 0, stream>>>(x, W_enc, b_enc, pre);

    const size_t smem = (size_t)D_SAE * 4 + 256 * 4 + 4 * 4;   // ~66.6KB << 320KB
    sae_topk<<<N_TOKENS, 256, smem, stream>>>(pre, sparse, tidx, tval);
    sae_decoder<<<N_TOKENS, 256, 0, stream>>>(WdecT, b_dec, tidx, tval, recon);
}